// SO3onS2_13675175871079
// MI455X (gfx1250) — compile-verified
//
#include <hip/hip_runtime.h>
#include <math.h>

#define LMAXV 8
#define NM 9            // LMAX+1
#define NLATV 128
#define NLONV 256
#define BATCHV 2048
#define NOUTV 215
#define MTOT (BATCHV * NLATV)   // 262144 GEMM rows
#define XHC 18                  // stored complex-interleaved cols of xh
#define CGSTRIDE 289            // max (2*l1+1)*(2*l2+1)

typedef __attribute__((ext_vector_type(2))) float v2f;
typedef __attribute__((ext_vector_type(8))) float v8f;

// ---------- index map: oi -> (l1, l2, lv), matching reference enumeration ----------
__device__ __forceinline__ void oi_to_lll(int oi, int& l1, int& l2, int& lv) {
  int c = 0;
  for (int a = 0; a <= LMAXV; ++a) {
    for (int b = a; b <= LMAXV; ++b) {
      int lo = b - a;
      int hi = min(a + b, LMAXV);
      int cnt = hi - lo + 1;
      if (oi < c + cnt) { l1 = a; l2 = b; lv = lo + (oi - c); return; }
      c += cnt;
    }
  }
  l1 = l2 = lv = 0;
}

// ---------- init: scaled DFT basis, pair-interleaved for b64 B-fragment loads ----------
// dftp[e*64 + n*2 + j] = B[2e+j][n],  where
//   B[s][2m]   =  cos(2*pi*m*s/256) * (2*pi/256)
//   B[s][2m+1] = -sin(2*pi*m*s/256) * (2*pi/256)   (rfft convention), cols 18..31 = 0
__global__ void init_dftp(float* __restrict__ dftp) {
  int e = blockIdx.x;       // 0..127 (even-sample pair index)
  int tid = threadIdx.x;    // 0..63
  int n = tid >> 1;
  int j = tid & 1;
  int s = 2 * e + j;
  float v = 0.f;
  if (n < 2 * NM) {
    int m = n >> 1;
    double ang = 2.0 * M_PI * (double)(m * s) / (double)NLONV;
    double sc = 2.0 * M_PI / (double)NLONV;
    v = (n & 1) ? (float)(-sin(ang) * sc) : (float)(cos(ang) * sc);
  }
  dftp[e * 64 + n * 2 + j] = v;
}

// ---------- init: orthonormal assoc. Legendre * quadrature weight ----------
__global__ void init_leg(float* __restrict__ legw) {
  int t = threadIdx.x;  // 0..127
  double theta = ((double)t + 0.5) * M_PI / (double)NLATV;
  double x = cos(theta), s = sin(theta);
  double w = (M_PI / (double)NLATV) * s;
  double P[NM][NM];
  for (int l = 0; l < NM; ++l)
    for (int m = 0; m < NM; ++m) P[l][m] = 0.0;
  P[0][0] = sqrt(1.0 / (4.0 * M_PI));
  for (int m = 1; m <= LMAXV; ++m)
    P[m][m] = -sqrt((2.0 * m + 1.0) / (2.0 * m)) * s * P[m - 1][m - 1];
  for (int m = 0; m <= LMAXV; ++m) {
    if (m + 1 <= LMAXV) P[m + 1][m] = sqrt(2.0 * m + 3.0) * x * P[m][m];
    for (int l = m + 2; l <= LMAXV; ++l) {
      double dl = (double)l, dm = (double)m;
      double a = sqrt((4.0 * dl * dl - 1.0) / (dl * dl - dm * dm));
      double b = sqrt(((dl - 1.0) * (dl - 1.0) - dm * dm) /
                      (4.0 * (dl - 1.0) * (dl - 1.0) - 1.0));
      P[l][m] = a * (x * P[l - 1][m] - b * P[l - 2][m]);
    }
  }
  for (int l = 0; l < NM; ++l)
    for (int m = 0; m < NM; ++m)
      legw[(l * NM + m) * NLATV + t] = (float)(P[l][m] * w);
}

// ---------- init: Clebsch-Gordan tables, one row block per output ----------
__device__ __forceinline__ double factd(int n) {
  double r = 1.0;
  for (int i = 2; i <= n; ++i) r *= (double)i;
  return r;
}

__global__ void init_cg(float* __restrict__ cgt) {
  int oi = blockIdx.x;
  int tid = threadIdx.x;
  int l1, l2, lv;
  oi_to_lll(oi, l1, l2, lv);
  int n2 = 2 * l2 + 1;
  int rows = (2 * l1 + 1) * n2;
  if (tid >= rows) return;
  int i1 = tid / n2, i2 = tid % n2;
  int m1 = i1 - l1, m2 = i2 - l2, M = m1 + m2;
  float val = 0.f;
  if (M >= -lv && M <= lv) {
    double pref = sqrt((2.0 * lv + 1.0) * factd(lv + l1 - l2) * factd(lv - l1 + l2) *
                       factd(l1 + l2 - lv) / factd(l1 + l2 + lv + 1));
    pref *= sqrt(factd(lv + M) * factd(lv - M) * factd(l1 - m1) * factd(l1 + m1) *
                 factd(l2 - m2) * factd(l2 + m2));
    int klo = max(0, max(l2 - lv - m1, l1 + m2 - lv));
    int khi = min(l1 + l2 - lv, min(l1 - m1, l2 + m2));
    double ssum = 0.0;
    for (int k = klo; k <= khi; ++k) {
      double d = factd(k) * factd(l1 + l2 - lv - k) * factd(l1 - m1 - k) *
                 factd(l2 + m2 - k) * factd(lv - l2 + m1 + k) * factd(lv - l1 - m2 + k);
      ssum += ((k & 1) ? -1.0 : 1.0) / d;
    }
    val = (float)(pref * ssum);
  }
  cgt[oi * CGSTRIDE + tid] = val;
}

// ---------- stage 1: DFT projection GEMM via V_WMMA_F32_16X16X4_F32 ----------
// A = x (262144 x 256) row-major fp32, B = dftp (pair-interleaved 128 x 32 x 2),
// D -> xh (262144 x 18). Each wave: 32-row M-tile x 32 cols, 4 accumulators.
__global__ void __launch_bounds__(256) dft_wmma(const float* __restrict__ x,
                                                const float* __restrict__ dftp,
                                                float* __restrict__ xh) {
  int wave = threadIdx.x >> 5;
  int lane = threadIdx.x & 31;
  int base = (blockIdx.x * 8 + wave) * 32;  // first of 32 rows for this wave
  int lr = lane & 15;
  int hi = lane >> 4;                        // 0 for lanes 0-15, 1 for lanes 16-31
  int ksub = hi << 1;                        // A fragment K sub-offset: 0 or 2
  const float* ap0 = x + (size_t)(base + lr) * NLONV + ksub;
  const float* ap1 = x + (size_t)(base + 16 + lr) * NLONV + ksub;
  // B fragment base: pair index e = k/2 + hi, column = lr (+16 for second N-tile)
  const float* bp = dftp + hi * 64 + lr * 2;
  v8f c00 = {0.f, 0.f, 0.f, 0.f, 0.f, 0.f, 0.f, 0.f};
  v8f c01 = c00, c10 = c00, c11 = c00;
  for (int k = 0; k < NLONV; k += 4) {
    v2f a0 = __builtin_nontemporal_load((const v2f*)(ap0 + k));
    v2f a1 = __builtin_nontemporal_load((const v2f*)(ap1 + k));
    v2f b0 = *(const v2f*)(bp + k * 32);        // (k/2)*64
    v2f b1 = *(const v2f*)(bp + k * 32 + 32);   // second N-tile (cols 16..31)
    c00 = __builtin_amdgcn_wmma_f32_16x16x4_f32(false, a0, false, b0, (short)0, c00,
                                                false, false);
    c01 = __builtin_amdgcn_wmma_f32_16x16x4_f32(false, a0, false, b1, (short)0, c01,
                                                false, false);
    c10 = __builtin_amdgcn_wmma_f32_16x16x4_f32(false, a1, false, b0, (short)0, c10,
                                                false, false);
    c11 = __builtin_amdgcn_wmma_f32_16x16x4_f32(false, a1, false, b1, (short)0, c11,
                                                false, false);
  }
  // D layout: VGPR i -> row i (lanes 0-15) / row i+8 (lanes 16-31), col = n0 + (lane&15)
  int rbase = base + (hi << 3);
  int col = lr;
#pragma unroll
  for (int i = 0; i < 8; ++i) {
    int r0 = rbase + i;
    int r1 = rbase + 16 + i;
    xh[(size_t)r0 * XHC + col] = c00[i];  // cols 0..15
    xh[(size_t)r1 * XHC + col] = c10[i];
    if (col + 16 < XHC) {                 // cols 16,17 only
      xh[(size_t)r0 * XHC + 16 + col] = c01[i];
      xh[(size_t)r1 * XHC + 16 + col] = c11[i];
    }
  }
}

// ---------- stage 2: Legendre projection  coeffs[b,l,m] = sum_t legw[l,m,t]*xh[b,t,m] ----------
__global__ void __launch_bounds__(96) leg_proj(const float* __restrict__ xh,
                                               const float* __restrict__ legw,
                                               float* __restrict__ coeffs) {
  int b = blockIdx.x;
  int tid = threadIdx.x;
  if (tid >= NM * NM) return;
  int l = tid / NM, m = tid % NM;
  const float* wv = legw + (l * NM + m) * NLATV;
  const float* xp = xh + (size_t)b * NLATV * XHC + 2 * m;
  float ar = 0.f, ai = 0.f;
  for (int t = 0; t < NLATV; ++t) {
    float w = wv[t];
    v2f z = *(const v2f*)(xp + t * XHC);  // (re, im) contiguous -> one b64 load
    ar += w * z.x;
    ai += w * z.y;
  }
  coeffs[((size_t)b * NM * NM + tid) * 2 + 0] = ar;
  coeffs[((size_t)b * NM * NM + tid) * 2 + 1] = ai;
}

// ---------- stage 3: bispectrum with CG selection-rule sparsity ----------
__global__ void __launch_bounds__(224) bispec(const float* __restrict__ coeffs,
                                              const float* __restrict__ cgt,
                                              float* __restrict__ out) {
  int b = blockIdx.x;
  int tid = threadIdx.x;
  __shared__ float Fr[81];
  __shared__ float Fi[81];
  if (tid < 81) {
    int l = 0;
    while ((l + 1) * (l + 1) <= tid) ++l;   // F[l] lives at indices l^2 .. l^2+2l
    int m = (tid - l * l) - l;
    const float* cb = coeffs + (size_t)b * 81 * 2;
    float fr, fi;
    if (m >= 0) {
      fr = cb[(l * NM + m) * 2 + 0];
      fi = cb[(l * NM + m) * 2 + 1];
    } else {
      int mm = -m;
      float sgn = (mm & 1) ? -1.f : 1.f;
      fr = sgn * cb[(l * NM + mm) * 2 + 0];
      fi = -sgn * cb[(l * NM + mm) * 2 + 1];
    }
    Fr[tid] = fr;
    Fi[tid] = fi;
  }
  __syncthreads();
  if (tid >= NOUTV) return;
  int l1, l2, lv;
  oi_to_lll(tid, l1, l2, lv);
  int o1 = l1 * l1, o2 = l2 * l2, ov = lv * lv;
  int n2 = 2 * l2 + 1;
  const float* cgrow = cgt + tid * CGSTRIDE;
  float accr = 0.f, acci = 0.f;
  for (int i1 = 0; i1 <= 2 * l1; ++i1) {
    float f1r = Fr[o1 + i1], f1i = Fi[o1 + i1];
    int m1 = i1 - l1;
    for (int i2 = 0; i2 < n2; ++i2) {
      int M = m1 + i2 - l2;
      if (M < -lv || M > lv) continue;
      float c = cgrow[i1 * n2 + i2];
      float f2r = Fr[o2 + i2], f2i = Fi[o2 + i2];
      float pr = f1r * f2r - f1i * f2i;
      float pi = f1r * f2i + f1i * f2r;
      float vr = Fr[ov + M + lv], vi = Fi[ov + M + lv];
      accr += c * (pr * vr + pi * vi);   // (F1*F2) * conj(Flv)
      acci += c * (pi * vr - pr * vi);
    }
  }
  out[((size_t)b * NOUTV + tid) * 2 + 0] = accr;
  out[((size_t)b * NOUTV + tid) * 2 + 1] = acci;
}

extern "C" void kernel_launch(void* const* d_in, const int* in_sizes, int n_in,
                              void* d_out, int out_size, void* d_ws, size_t ws_size,
                              hipStream_t stream) {
  const float* x = (const float*)d_in[0];
  float* out = (float*)d_out;
  float* ws = (float*)d_ws;
  // workspace layout (float offsets)
  float* dftp = ws;                                 // 128*64          = 8192
  float* legw = ws + 8192;                          // 81*128          = 10368
  float* cgt = ws + 18560;                          // 215*289         = 62135
  float* xh = ws + 81920;                           // 262144*18       = 4718592
  float* coeffs = ws + 81920 + (size_t)MTOT * XHC;  // 2048*81*2       = 331776
  // ~20.5 MB total workspace

  hipLaunchKernelGGL(init_dftp, dim3(128), dim3(64), 0, stream, dftp);
  hipLaunchKernelGGL(init_leg, dim3(1), dim3(NLATV), 0, stream, legw);
  hipLaunchKernelGGL(init_cg, dim3(NOUTV), dim3(320), 0, stream, cgt);
  hipLaunchKernelGGL(dft_wmma, dim3(MTOT / 256), dim3(256), 0, stream, x, dftp, xh);
  hipLaunchKernelGGL(leg_proj, dim3(BATCHV), dim3(96), 0, stream, xh, legw, coeffs);
  hipLaunchKernelGGL(bispec, dim3(BATCHV), dim3(224), 0, stream, coeffs, cgt, out);
}